// MeanPoolDynamicChunker_63788854280488
// MI455X (gfx1250) — compile-verified
//
#include <hip/hip_runtime.h>
#include <hip/hip_bf16.h>

// ---------------------------------------------------------------------------
// MeanPoolDynamicChunker on MI455X (gfx1250, wave32).
//
// Pipeline (all on `stream`):
//   k_valid   : vf[b,t] = (sum_d |frames|) != 0
//   k_cumsum  : csf[b,t,d] prefix sums over t; csv[b,t] prefix sums of vf
//   k_flags   : dflag[r] = (denom==0) for each window row r=(b,i,l)
//   k_transpose: Wt* = W*^T ([n][k] layout) so B-fragments are one b64 load
//   k_gemm12  : fused  mean -> H1(LDS, 48KB) -> rep(global), f32 WMMA 16x16x4
//   k_gemm3   : fused  rep@W3 + relu + (*w4) row-reduction -> is_end[r]
//   k_scan    : sequential per-batch boundary scan + rep gather -> d_out
//
// Precision: full f32 via V_WMMA_F32_16X16X4_F32 (the sigmoid threshold is
// p>0.5 <=> logit>0 with logits near zero; low-precision inputs would flip
// chunk boundaries). K=4 per WMMA makes the loop issue-bound, so we minimize
// VMEM issues per WMMA: transposed weights (1x b64 per B-frag) and paired
// column tiles (each A-frag feeds two WMMAs).
// ---------------------------------------------------------------------------

typedef float f2  __attribute__((ext_vector_type(2)));
typedef float v8f __attribute__((ext_vector_type(8)));

#define Bb   4
#define Tt   512
#define Dd   768
#define Ll   32
#define TP1  513            // T+1
#define RR   (Bb*Tt*Ll)     // 65536 candidate-window rows

// ---------------------------------------------------------------- validity
__global__ void k_valid(const float* __restrict__ frames, float* __restrict__ vf) {
    int bt = blockIdx.x;                       // 0 .. B*T-1
    __shared__ float red[256];
    float s = 0.f;
    for (int c = threadIdx.x; c < Dd; c += 256)
        s += fabsf(frames[(long)bt * Dd + c]);
    red[threadIdx.x] = s;
    __syncthreads();
    for (int w = 128; w > 0; w >>= 1) {
        if (threadIdx.x < w) red[threadIdx.x] += red[threadIdx.x + w];
        __syncthreads();
    }
    if (threadIdx.x == 0) vf[bt] = (red[0] != 0.f) ? 1.f : 0.f;
}

// ---------------------------------------------------------------- cumsums
__global__ void k_cumsum(const float* __restrict__ frames, const float* __restrict__ vf,
                         float* __restrict__ csf, float* __restrict__ csv) {
    int gid = blockIdx.x * blockDim.x + threadIdx.x;
    if (gid < Bb * Dd) {                       // csf: one (b,d) column each
        int b = gid / Dd, d = gid % Dd;
        float acc = 0.f;
        csf[((long)(b * TP1) + 0) * Dd + d] = 0.f;
        for (int t = 0; t < Tt; ++t) {
            acc += frames[((long)(b * Tt) + t) * Dd + d];
            csf[((long)(b * TP1) + t + 1) * Dd + d] = acc;
        }
    }
    if (gid < Bb) {                            // csv: one b each
        int b = gid;
        float acc = 0.f;
        csv[b * TP1] = 0.f;
        for (int t = 0; t < Tt; ++t) {
            acc += vf[b * Tt + t];
            csv[b * TP1 + t + 1] = acc;
        }
    }
}

// ---------------------------------------------------------------- denom flags
__global__ void k_flags(const float* __restrict__ csv, int* __restrict__ dflag) {
    int r = blockIdx.x * blockDim.x + threadIdx.x;
    if (r >= RR) return;
    int b = r / (Tt * Ll);
    int rem = r % (Tt * Ll);
    int i = rem / Ll, l = rem % Ll;
    int e = i + l + 1; if (e > Tt) e = Tt;
    float denom = csv[b * TP1 + e] - csv[b * TP1 + i];
    dflag[r] = (denom == 0.f) ? 1 : 0;
}

// ---------------------------------------------------------------- weight transpose
__global__ void k_transpose(const float* __restrict__ W1, const float* __restrict__ W2,
                            const float* __restrict__ W3,
                            float* __restrict__ Wt1, float* __restrict__ Wt2,
                            float* __restrict__ Wt3) {
    int idx = blockIdx.x * blockDim.x + threadIdx.x;   // over D*D
    if (idx >= Dd * Dd) return;
    int k = idx / Dd, n = idx % Dd;
    Wt1[n * Dd + k] = W1[idx];
    Wt2[n * Dd + k] = W2[idx];
    Wt3[n * Dd + k] = W3[idx];
}

// ---------------------------------------------------------------- fused GEMM1+GEMM2
// Block = 16 rows of the 65536-row window matrix. 4 waves, each owns 12 of the
// 48 column tiles (processed as 6 pairs). H1 stays in LDS.
__global__ __launch_bounds__(128) void
k_gemm12(const float* __restrict__ csf, const float* __restrict__ csv,
         const int* __restrict__ dflag,
         const float* __restrict__ Wt1, const float* __restrict__ b1,
         const float* __restrict__ Wt2, const float* __restrict__ b2,
         float* __restrict__ rep) {
    __shared__ float H1s[16 * Dd];             // 48 KB
    __shared__ int   dsh[16];

    int lane = threadIdx.x & 31;
    int wave = threadIdx.x >> 5;
    int m    = lane & 15;                      // A-matrix row held by this lane
    int kOff = (lane >> 4) << 1;               // upper half-wave holds K+2,K+3
    int r0   = blockIdx.x * 16;
    int r    = r0 + m;

    int b = r / (Tt * Ll);
    int rem = r % (Tt * Ll);
    int i = rem / Ll, l = rem & (Ll - 1);
    int e = i + l + 1; if (e > Tt) e = Tt;
    float denom = csv[b * TP1 + e] - csv[b * TP1 + i];
    float invd  = 1.f / fmaxf(denom, 1.f);

    if (threadIdx.x < 16) dsh[threadIdx.x] = dflag[r0 + threadIdx.x];

    const float* pE = csf + ((long)(b * TP1 + e)) * Dd;
    const float* pI = csf + ((long)(b * TP1 + i)) * Dd;

    // ---- GEMM1: H1 = relu(mean @ W1 + b1), A synthesized from csf ----
    for (int p = 0; p < 6; ++p) {
        int n0 = (wave * 12 + 2 * p) * 16;
        int nA = n0 + m, nB = n0 + 16 + m;
        const float* wc0 = Wt1 + (long)nA * Dd;
        const float* wc1 = Wt1 + (long)nB * Dd;
        v8f d0 = {}, d1 = {};
        #pragma unroll 4
        for (int k0 = 0; k0 < Dd; k0 += 4) {
            int kA = k0 + kOff;
            f2 ae = *(const f2*)(pE + kA);
            f2 ai = *(const f2*)(pI + kA);
            f2 a  = (ae - ai) * invd;          // masked window mean, on the fly
            f2 bf0 = *(const f2*)(wc0 + kA);   // transposed: contiguous b64
            f2 bf1 = *(const f2*)(wc1 + kA);
            d0 = __builtin_amdgcn_wmma_f32_16x16x4_f32(false, a, false, bf0,
                                                       (short)0, d0, false, false);
            d1 = __builtin_amdgcn_wmma_f32_16x16x4_f32(false, a, false, bf1,
                                                       (short)0, d1, false, false);
        }
        float biasA = b1[nA], biasB = b1[nB];
        #pragma unroll
        for (int j = 0; j < 8; ++j) {
            int row = (lane < 16) ? j : j + 8;
            H1s[row * Dd + nA] = fmaxf(d0[j] + biasA, 0.f);
            H1s[row * Dd + nB] = fmaxf(d1[j] + biasB, 0.f);
        }
    }
    __syncthreads();

    // ---- GEMM2: rep = H1 @ W2 + b2 (zero rows where denom==0) ----
    for (int p = 0; p < 6; ++p) {
        int n0 = (wave * 12 + 2 * p) * 16;
        int nA = n0 + m, nB = n0 + 16 + m;
        const float* wc0 = Wt2 + (long)nA * Dd;
        const float* wc1 = Wt2 + (long)nB * Dd;
        v8f d0 = {}, d1 = {};
        #pragma unroll 4
        for (int k0 = 0; k0 < Dd; k0 += 4) {
            int kA = k0 + kOff;
            f2 a = *(const f2*)(&H1s[m * Dd + kA]);
            f2 bf0 = *(const f2*)(wc0 + kA);
            f2 bf1 = *(const f2*)(wc1 + kA);
            d0 = __builtin_amdgcn_wmma_f32_16x16x4_f32(false, a, false, bf0,
                                                       (short)0, d0, false, false);
            d1 = __builtin_amdgcn_wmma_f32_16x16x4_f32(false, a, false, bf1,
                                                       (short)0, d1, false, false);
        }
        float biasA = b2[nA], biasB = b2[nB];
        #pragma unroll
        for (int j = 0; j < 8; ++j) {
            int row = (lane < 16) ? j : j + 8;
            float vA = d0[j] + biasA;
            float vB = d1[j] + biasB;
            if (dsh[row]) { vA = 0.f; vB = 0.f; }
            rep[((long)(r0 + row)) * Dd + nA] = vA;
            rep[((long)(r0 + row)) * Dd + nB] = vB;
        }
    }
}

// ---------------------------------------------------------------- fused GEMM3 + eoc head
// logit[r] = sum_n relu((rep@W3)[r,n] + b3[n]) * w4[n] + b4  -> is_end[r]
__global__ __launch_bounds__(128) void
k_gemm3(const float* __restrict__ rep,
        const float* __restrict__ Wt3, const float* __restrict__ b3,
        const float* __restrict__ w4, const float* __restrict__ b4,
        const int* __restrict__ dflag, int* __restrict__ is_end) {
    int lane = threadIdx.x & 31;
    int wave = threadIdx.x >> 5;
    int m    = lane & 15;
    int kOff = (lane >> 4) << 1;
    int r0   = (blockIdx.x * 4 + wave) * 16;   // each wave owns 16 rows

    const float* pA = rep + ((long)(r0 + m)) * Dd;
    float acc[8] = {0.f, 0.f, 0.f, 0.f, 0.f, 0.f, 0.f, 0.f};

    for (int p = 0; p < 24; ++p) {             // 48 column tiles, in pairs
        int n0 = p * 32;
        int nA = n0 + m, nB = n0 + 16 + m;
        const float* wc0 = Wt3 + (long)nA * Dd;
        const float* wc1 = Wt3 + (long)nB * Dd;
        v8f d0 = {}, d1 = {};
        #pragma unroll 4
        for (int k0 = 0; k0 < Dd; k0 += 4) {
            int kA = k0 + kOff;
            f2 a = *(const f2*)(pA + kA);
            f2 bf0 = *(const f2*)(wc0 + kA);
            f2 bf1 = *(const f2*)(wc1 + kA);
            d0 = __builtin_amdgcn_wmma_f32_16x16x4_f32(false, a, false, bf0,
                                                       (short)0, d0, false, false);
            d1 = __builtin_amdgcn_wmma_f32_16x16x4_f32(false, a, false, bf1,
                                                       (short)0, d1, false, false);
        }
        float b3A = b3[nA], w4A = w4[nA];
        float b3B = b3[nB], w4B = w4[nB];
        #pragma unroll
        for (int j = 0; j < 8; ++j) {
            acc[j] += fmaxf(d0[j] + b3A, 0.f) * w4A;
            acc[j] += fmaxf(d1[j] + b3B, 0.f) * w4B;
        }
    }

    // reduce the 16 column-lanes within each half-wave (rows j / j+8)
    #pragma unroll
    for (int j = 0; j < 8; ++j) {
        acc[j] += __shfl_xor(acc[j], 1, 32);
        acc[j] += __shfl_xor(acc[j], 2, 32);
        acc[j] += __shfl_xor(acc[j], 4, 32);
        acc[j] += __shfl_xor(acc[j], 8, 32);
    }
    if (m == 0) {                              // lanes 0 and 16 hold row sums
        float b4v = b4[0];
        #pragma unroll
        for (int j = 0; j < 8; ++j) {
            int row = (lane < 16) ? j : j + 8;
            int rr  = r0 + row;
            float logit = acc[j] + b4v;        // p>0.5  <=>  logit>0
            is_end[rr] = ((logit > 0.f) || dflag[rr]) ? 1 : 0;
        }
    }
}

// ---------------------------------------------------------------- boundary scan + gather
__global__ void k_scan(const float* __restrict__ rep, const int* __restrict__ is_end,
                       const float* __restrict__ csv, float* __restrict__ out) {
    int b = blockIdx.x;                        // one block per batch sample
    __shared__ int sh_row, sh_active;
    float totv = csv[b * TP1 + Tt];
    int i = 0, cnt = 0;
    for (int t = 0; t < Tt; ++t) {
        if (threadIdx.x == 0) {
            int ic = (i < Tt - 1) ? i : Tt - 1;
            bool suf    = (totv - csv[b * TP1 + ic]) > 0.f;
            bool active = (i < Tt) && suf;
            int lmax = Tt - ic; if (lmax > Ll) lmax = Ll;
            int lstar = lmax;
            for (int l = 0; l < lmax; ++l) {
                if (is_end[(b * Tt + ic) * Ll + l]) { lstar = l + 1; break; }
            }
            sh_row    = (b * Tt + ic) * Ll + (lstar - 1);
            sh_active = active ? 1 : 0;
            if (active) { cnt++; i += lstar; }
        }
        __syncthreads();
        int act = sh_active;
        const float* src = rep + (long)sh_row * Dd;
        float* dst = out + ((long)(b * Tt + t)) * Dd;
        for (int c = threadIdx.x; c < Dd; c += blockDim.x)
            dst[c] = act ? src[c] : 0.f;
        __syncthreads();
    }
    if (threadIdx.x == 0)
        out[(long)Bb * Tt * Dd + b] = (float)(cnt > 0 ? cnt : 1);
}

// ---------------------------------------------------------------- launch
extern "C" void kernel_launch(void* const* d_in, const int* in_sizes, int n_in,
                              void* d_out, int out_size, void* d_ws, size_t ws_size,
                              hipStream_t stream) {
    const float* frames = (const float*)d_in[0];
    const float* W1 = (const float*)d_in[1];
    const float* b1 = (const float*)d_in[2];
    const float* W2 = (const float*)d_in[3];
    const float* b2 = (const float*)d_in[4];
    const float* W3 = (const float*)d_in[5];
    const float* b3 = (const float*)d_in[6];
    const float* w4 = (const float*)d_in[7];
    const float* b4 = (const float*)d_in[8];

    // workspace carve-up (all 16B aligned)
    float* csf = (float*)d_ws;                         // B*513*D
    float* csv = csf + (size_t)Bb * TP1 * Dd;          // B*513
    float* vf  = csv + 2052;                           // B*T (2052 keeps 16B align)
    float* rep = vf + 2048;                            // R*D
    int* dflag = (int*)(rep + (size_t)RR * Dd);        // R
    int* is_end = dflag + RR;                          // R
    float* Wt1 = (float*)(is_end + RR);                // D*D each, transposed
    float* Wt2 = Wt1 + (size_t)Dd * Dd;
    float* Wt3 = Wt2 + (size_t)Dd * Dd;

    k_valid    <<<Bb * Tt, 256, 0, stream>>>(frames, vf);
    k_cumsum   <<<12, 256, 0, stream>>>(frames, vf, csf, csv);
    k_flags    <<<RR / 256, 256, 0, stream>>>(csv, dflag);
    k_transpose<<<(Dd * Dd + 255) / 256, 256, 0, stream>>>(W1, W2, W3, Wt1, Wt2, Wt3);
    k_gemm12   <<<RR / 16, 128, 0, stream>>>(csf, csv, dflag, Wt1, b1, Wt2, b2, rep);
    k_gemm3    <<<RR / 64, 128, 0, stream>>>(rep, Wt3, b3, w4, b4, dflag, is_end);
    k_scan     <<<Bb, 256, 0, stream>>>(rep, is_end, csv, (float*)d_out);
}